// EIGNN_w_iterative_solvers_52733608461011
// MI455X (gfx1250) — compile-verified
//
#include <hip/hip_runtime.h>
#include <math.h>

#define GAMMA   0.8f
#define EPS_F   1e-12f
#define MDIM    256
#define APPLY   301      // 300 loop iterations + 1 final inner() application
#define PSTRIDE 544      // floats per interleaved k-pair row (512 data + 32 pad -> bank-conflict-free b64)

typedef float v2f __attribute__((ext_vector_type(2)));
typedef float v8f __attribute__((ext_vector_type(8)));
typedef unsigned int u32;

// ---------------- FF = F^T F  (256x256), plus Frobenius sum-of-squares ----------------
__global__ void k_ff_norm(const float* __restrict__ F, float* __restrict__ FF,
                          float* __restrict__ normAcc) {
  const int i = blockIdx.x;     // FF row
  const int j = threadIdx.x;    // FF col
  float acc = 0.f;
#pragma unroll 4
  for (int k = 0; k < MDIM; ++k)
    acc = fmaf(F[k * MDIM + i], F[k * MDIM + j], acc);
  FF[i * MDIM + j] = acc;
  __shared__ float red[MDIM];
  red[j] = acc * acc;
  __syncthreads();
  for (int off = 128; off > 0; off >>= 1) {
    if (j < off) red[j] += red[j + off];
    __syncthreads();
  }
  if (j == 0) atomicAdd(normAcc, red[0]);
}

// GTi = GAMMA * FF / (||FF||_F + eps), stored k-pair interleaved:
// GTi[(k>>1)*PSTRIDE + m*2 + (k&1)] = scaled FF[k][m]   (FF symmetric -> also (gamma*GF)^T)
__global__ void k_scale_gf(const float* __restrict__ FF, const float* __restrict__ normAcc,
                           float* __restrict__ GTi) {
  const int k = blockIdx.x, m = threadIdx.x;
  const float s = GAMMA / (sqrtf(*normAcc) + EPS_F);
  GTi[(size_t)(k >> 1) * PSTRIDE + m * 2 + (k & 1)] = FF[k * MDIM + m] * s;
}

// ---------------- CSC build ----------------
__global__ void k_hist(const int* __restrict__ cols, u32* __restrict__ cnt, int nnz) {
  const int e = blockIdx.x * blockDim.x + threadIdx.x;
  if (e < nnz) atomicAdd(&cnt[cols[e]], 1u);
}

// single-block exclusive scan over N counts -> colPtr[0..N]; cursor[i] = colPtr[i]
__global__ void k_scan(u32* __restrict__ cursor, u32* __restrict__ colPtr, int n) {
  __shared__ u32 s[1024];
  const int tid = threadIdx.x;
  u32 running = 0;
  for (int base = 0; base < n; base += 1024) {
    const int i = base + tid;
    const u32 v = (i < n) ? cursor[i] : 0u;
    s[tid] = v;
    __syncthreads();
    for (int off = 1; off < 1024; off <<= 1) {
      u32 t = (tid >= off) ? s[tid - off] : 0u;
      __syncthreads();
      s[tid] += t;
      __syncthreads();
    }
    const u32 excl = running + s[tid] - v;
    if (i < n) { colPtr[i] = excl; cursor[i] = excl; }
    const u32 tot = s[1023];
    __syncthreads();
    running += tot;
  }
  if (tid == 0) colPtr[n] = running;
}

__global__ void k_scatter(const int* __restrict__ rows, const int* __restrict__ cols,
                          const float* __restrict__ vals, u32* __restrict__ cursor,
                          int* __restrict__ cscRow, float* __restrict__ cscVal, int nnz) {
  const int e = blockIdx.x * blockDim.x + threadIdx.x;
  if (e < nnz) {
    const int c = cols[e];
    const u32 p = atomicAdd(&cursor[c], 1u);
    cscRow[p] = rows[e];
    cscVal[p] = vals[e];
  }
}

// ---------------- transposes (X[M,N] -> Xt[Npad,M], Zt[Npad,M] -> out[M,N]) ----------------
__global__ void k_xpose_in(const float* __restrict__ X, float* __restrict__ Xt, int n) {
  __shared__ float tile[32][33];
  const int tx = threadIdx.x, ty = threadIdx.y;
  const int n0 = blockIdx.x * 32, m0 = blockIdx.y * 32;
#pragma unroll
  for (int i = 0; i < 4; ++i) {
    const int nn = n0 + tx;
    tile[ty + i * 8][tx] = (nn < n) ? X[(size_t)(m0 + ty + i * 8) * n + nn] : 0.f;
  }
  __syncthreads();
#pragma unroll
  for (int i = 0; i < 4; ++i) {
    const int nn = n0 + ty + i * 8;
    if (nn < n) Xt[(size_t)nn * MDIM + m0 + tx] = tile[tx][ty + i * 8];
  }
}

__global__ void k_xpose_out(const float* __restrict__ Zt, float* __restrict__ out, int n) {
  __shared__ float tile[32][33];
  const int tx = threadIdx.x, ty = threadIdx.y;
  const int n0 = blockIdx.x * 32, m0 = blockIdx.y * 32;
#pragma unroll
  for (int i = 0; i < 4; ++i)
    tile[ty + i * 8][tx] = Zt[(size_t)(n0 + ty + i * 8) * MDIM + m0 + tx]; // Npad-safe
  __syncthreads();
#pragma unroll
  for (int i = 0; i < 4; ++i) {
    const int nn = n0 + tx;
    if (nn < n) out[(size_t)(m0 + ty + i * 8) * n + nn] = tile[tx][ty + i * 8];
  }
}

// ---------------- hot: SpMM gather  prop[c,:] = sum_e v_e * Zt[r_e,:] ----------------
__global__ __launch_bounds__(256) void k_spmm(const u32* __restrict__ colPtr,
                                              const int* __restrict__ cscRow,
                                              const float* __restrict__ cscVal,
                                              const float* __restrict__ Zt,
                                              float* __restrict__ prop, int n) {
  const int node = blockIdx.x * 4 + (threadIdx.x >> 6);
  const int q = threadIdx.x & 63;               // 64 lanes x float4 = 256 features
  if (node >= n) return;
  const int s = (int)colPtr[node];
  const int e = (int)colPtr[node + 1];
  const float4* __restrict__ Zt4 = (const float4*)Zt;
  float4 acc = make_float4(0.f, 0.f, 0.f, 0.f);
  for (int j = s; j < e; ++j) {
    const int r = cscRow[j];
    const float v = cscVal[j];
    const float4 z = Zt4[(size_t)r * 64 + q];   // contiguous 1KB per edge across lanes
    acc.x = fmaf(v, z.x, acc.x);
    acc.y = fmaf(v, z.y, acc.y);
    acc.z = fmaf(v, z.z, acc.z);
    acc.w = fmaf(v, z.w, acc.w);
  }
  ((float4*)prop)[(size_t)node * 64 + q] = acc;
}

// ---------------- hot: Znew_t[n,m] = sum_k P[n,k]*GT[k,m] + Xt[n,m]  (fp32 WMMA) ----------
// Wave tile: 16(n) x 64(m), 4 x v8f accumulators seeded with Xt (fuses +X).
// Block: 256 threads = 8 waves -> 32(n) x 256(m).
// B staged in LDS in k-pair-interleaved form so each lane's {GT[k][m],GT[k+1][m]} is one
// aligned ds_load_b64; PSTRIDE=544 puts the K+2/3 half 32 banks away -> conflict-free.
__global__ __launch_bounds__(256) void k_gemm(const float* __restrict__ P,
                                              const float* __restrict__ GTi,
                                              const float* __restrict__ Xt,
                                              float* __restrict__ Zn) {
  __shared__ float Bs[16 * PSTRIDE];            // 34.8KB: 16 k-pairs (K=32 slab)
  const int tid = threadIdx.x;
  const int wid = tid >> 5;
  const int lane = tid & 31;
  const int half = lane >> 4;                   // 0: K+0/1 , 1: K+2/3 (A/B dual-half layout)
  const int l = lane & 15;
  const int m0 = (wid & 3) * 64;
  const int n0 = blockIdx.x * 32 + (wid >> 2) * 16;

  v8f acc[4];
#pragma unroll
  for (int t = 0; t < 4; ++t) {                 // seed C with Xt tile (C/D VGPR layout)
    const int mt = m0 + t * 16 + l;
    const float* xp = Xt + (size_t)(n0 + half * 8) * MDIM + mt;
#pragma unroll
    for (int j = 0; j < 8; ++j) acc[t][j] = xp[(size_t)j * MDIM];
  }

  const float* aBase = P + (size_t)(n0 + l) * MDIM + half * 2;

  for (int k0 = 0; k0 < MDIM; k0 += 32) {
    __syncthreads();
    {   // straight float4 copy of the interleaved slab: 16*PSTRIDE floats / 256 threads
      const float4* g4 = (const float4*)(GTi + (size_t)(k0 >> 1) * PSTRIDE);
      float4* b4 = (float4*)Bs;
      for (int u = tid; u < (16 * PSTRIDE) / 4; u += 256) b4[u] = g4[u];
    }
    __syncthreads();
#pragma unroll
    for (int kk = 0; kk < 32; kk += 4) {
      const int k = k0 + kk;
      const v2f a = *(const v2f*)(aBase + k);   // A: P[n0+l][k+half*2 .. +1]
#pragma unroll
      for (int t = 0; t < 4; ++t) {
        const int mt = m0 + t * 16 + l;
        // B: {GT[k+half*2][mt], GT[k+half*2+1][mt]} -> one aligned ds_load_b64
        const v2f b = *(const v2f*)(&Bs[((kk >> 1) + half) * PSTRIDE + mt * 2]);
        acc[t] = __builtin_amdgcn_wmma_f32_16x16x4_f32(
            false, a, false, b, (short)0, acc[t], false, false);
      }
    }
  }

#pragma unroll
  for (int t = 0; t < 4; ++t) {
    const int mt = m0 + t * 16 + l;
    float* zp = Zn + (size_t)(n0 + half * 8) * MDIM + mt;
#pragma unroll
    for (int j = 0; j < 8; ++j) zp[(size_t)j * MDIM] = acc[t][j];
  }
}

// ---------------- launch ----------------
extern "C" void kernel_launch(void* const* d_in, const int* in_sizes, int n_in,
                              void* d_out, int out_size, void* d_ws, size_t ws_size,
                              hipStream_t stream) {
  const float* X    = (const float*)d_in[0];   // [256, N]
  const float* F    = (const float*)d_in[1];   // [256, 256]
  const float* vals = (const float*)d_in[2];   // [NNZ]
  const int*   rows = (const int*)d_in[3];     // [NNZ]
  const int*   cols = (const int*)d_in[4];     // [NNZ]

  const int N    = in_sizes[0] / MDIM;
  const int NNZ  = in_sizes[2];
  const int Npad = ((N + 31) / 32) * 32;

  // workspace bump allocator (256B aligned)
  char* base = (char*)d_ws;
  size_t off = 0;
  auto alloc = [&](size_t bytes) -> char* {
    char* p = base + off;
    off += (bytes + 255) & ~(size_t)255;
    return p;
  };
  float* GTi     = (float*)alloc((size_t)(MDIM / 2) * PSTRIDE * 4);  // interleaved, padded
  float* FF      = (float*)alloc((size_t)MDIM * MDIM * 4);
  float* normAcc = (float*)alloc(4);
  u32*   colPtr  = (u32*)alloc((size_t)(N + 1) * 4);
  u32*   cursor  = (u32*)alloc((size_t)N * 4);
  int*   cscRow  = (int*)alloc((size_t)NNZ * 4);
  float* cscVal  = (float*)alloc((size_t)NNZ * 4);
  const size_t matBytes = (size_t)Npad * MDIM * 4;
  float* Xt   = (float*)alloc(matBytes);
  float* prop = (float*)alloc(matBytes);
  float* ZtA  = (float*)alloc(matBytes);
  float* ZtB  = (float*)alloc(matBytes);
  float* Zt[2] = {ZtA, ZtB};

  // zero-init (also clears the Npad padding rows once; pads are never re-dirtied)
  hipMemsetAsync(normAcc, 0, 4, stream);
  hipMemsetAsync(cursor, 0, (size_t)N * 4, stream);
  hipMemsetAsync(GTi, 0, (size_t)(MDIM / 2) * PSTRIDE * 4, stream);  // pad lanes read 0
  hipMemsetAsync(Xt,   0, matBytes, stream);
  hipMemsetAsync(prop, 0, matBytes, stream);
  hipMemsetAsync(ZtA,  0, matBytes, stream);   // z0 = 0

  // GF prep
  k_ff_norm<<<MDIM, MDIM, 0, stream>>>(F, FF, normAcc);
  k_scale_gf<<<MDIM, MDIM, 0, stream>>>(FF, normAcc, GTi);

  // CSC build
  const int eBlocks = (NNZ + 255) / 256;
  k_hist<<<eBlocks, 256, 0, stream>>>(cols, cursor, NNZ);
  k_scan<<<1, 1024, 0, stream>>>(cursor, colPtr, N);
  k_scatter<<<eBlocks, 256, 0, stream>>>(rows, cols, vals, cursor, cscRow, cscVal, NNZ);

  // X -> Xt
  dim3 tb(32, 8);
  dim3 tg((N + 31) / 32, MDIM / 32);
  k_xpose_in<<<tg, tb, 0, stream>>>(X, Xt, N);

  // fixed-point loop: 300 iterations + 1 final application (see analysis)
  const int spmmBlocks = (N + 3) / 4;
  const int gemmBlocks = Npad / 32;
  for (int t = 1; t <= APPLY; ++t) {
    k_spmm<<<spmmBlocks, 256, 0, stream>>>(colPtr, cscRow, cscVal, Zt[(t - 1) & 1], prop, N);
    k_gemm<<<gemmBlocks, 256, 0, stream>>>(prop, GTi, Xt, Zt[t & 1]);
  }

  // APPLY is odd -> result lives in Zt[1]; transpose back to [M, N]
  k_xpose_out<<<tg, tb, 0, stream>>>(Zt[1], (float*)d_out, N);
}